// GNN_Encoder_69947837382767
// MI455X (gfx1250) — compile-verified
//
#include <hip/hip_runtime.h>
#include <hip/hip_bf16.h>

typedef __attribute__((ext_vector_type(2))) float v2f;
typedef __attribute__((ext_vector_type(8))) float v8f;

#define GCN_N   50000
#define GCN_E   800000
#define GCN_IN  128
#define GCN_H   128
#define GCN_OUT 64
#define GCN_K   128          // all GEMM inputs are 128 wide
#define LDS_STRIDE 132       // bank-conflict-free padding (see analysis)

// ---------------------------------------------------------------------------
// Degree / normalization:  dis[i] = rsqrt(1 + in_degree(i))
// ---------------------------------------------------------------------------
__global__ void deg_init_kernel(float* __restrict__ dis, int n) {
    int i = blockIdx.x * blockDim.x + threadIdx.x;
    if (i < n) dis[i] = 1.0f;   // self-loop contribution to degree
}

__global__ void deg_count_kernel(const int* __restrict__ dst, float* __restrict__ dis, int e) {
    int i = blockIdx.x * blockDim.x + threadIdx.x;
    if (i < e) atomicAdd(&dis[dst[i]], 1.0f);
}

__global__ void deg_rsqrt_kernel(float* __restrict__ dis, int n) {
    int i = blockIdx.x * blockDim.x + threadIdx.x;
    if (i < n) dis[i] = rsqrtf(dis[i]);
}

// ---------------------------------------------------------------------------
// Fused: h = (RELU ? relu(A) : A) @ W ;  agg = h * dis[row]^2 + bias[col]
// A:[M,128] row-major, W:[128,FOUT] row-major.
// Block = FOUT/16 waves; blockIdx.x owns a 16-row tile, wave w owns 16-col
// tile w. The 16x128 A tile (8 KB) is staged once per block into LDS with
// ReLU applied, then each wave runs 32 chained V_WMMA_F32_16X16X4_F32
// (exact fp32), reading A fragments via conflict-free ds_load_b64.
// A-row r is read only by block r/16 (staging phase) and agg-row r is written
// only by that block (epilogue), so agg_out may alias A in-place: the
// __syncthreads() after staging orders all global A reads before any write.
// Grid/block are exact, every wave fully active -> EXEC all-ones at WMMA.
// ---------------------------------------------------------------------------
template <int FOUT, bool RELU>
__global__ void gemm_wmma_fused_kernel(const float* __restrict__ A,
                                       const float* __restrict__ W,
                                       const float* __restrict__ bias,
                                       const float* __restrict__ dis,
                                       float* __restrict__ h_out,
                                       float* __restrict__ agg_out) {
    constexpr int NTHREADS = 32 * (FOUT / 16);
    __shared__ float As[16 * LDS_STRIDE];

    const int tid  = threadIdx.x;
    const int wave = tid >> 5;
    const int lane = tid & 31;
    const int n0   = wave * 16;
    const int m0   = blockIdx.x * 16;

    // ---- stage A tile (16 x 128 f32) into LDS, ReLU fused, b128 loads ----
    #pragma unroll
    for (int idx = tid * 4; idx < 16 * GCN_K; idx += NTHREADS * 4) {
        const int r  = idx >> 7;      // / 128
        const int cc = idx & 127;
        float4 v = *(const float4*)(A + (size_t)(m0 + r) * GCN_K + cc);
        if (RELU) {
            v.x = fmaxf(v.x, 0.0f); v.y = fmaxf(v.y, 0.0f);
            v.z = fmaxf(v.z, 0.0f); v.w = fmaxf(v.w, 0.0f);
        }
        *(float4*)(&As[r * LDS_STRIDE + cc]) = v;
    }
    __syncthreads();   // also orders all global A reads before epilogue writes

    const int m  = lane & 15;   // A row / B,C column within tile
    const int kh = lane >> 4;   // K-half selector (0 or 1)

    v8f c = {};
    #pragma unroll
    for (int k = 0; k < GCN_K; k += 4) {
        // A 16x4 fragment from LDS: lane (m,kh) -> A[m][k+2kh], A[m][k+2kh+1]
        // ds_load_b64, conflict-free under LDS_STRIDE=132
        v2f a = *(const v2f*)(&As[m * LDS_STRIDE + k + 2 * kh]);
        // B 4x16 fragment: VGPR v holds W[k + v + 2kh][n0 + (lane&15)]
        v2f b;
        b.x = W[(size_t)(k + 2 * kh)     * FOUT + n0 + m];
        b.y = W[(size_t)(k + 2 * kh + 1) * FOUT + n0 + m];
        // 8 args: (neg_a, A, neg_b, B, c_mod, C, reuse_a, reuse_b)
        c = __builtin_amdgcn_wmma_f32_16x16x4_f32(false, a, false, b,
                                                  (short)0, c, false, false);
    }

    // ---- epilogue: write h and agg = h*dis^2 + bias (self-loop init) ----
    const int   col = n0 + m;
    const float bv  = bias[col];
    // C/D layout: VGPR i -> row m0 + i + 8*kh, col n0 + (lane&15)
    #pragma unroll
    for (int i = 0; i < 8; ++i) {
        const int   row = m0 + i + 8 * kh;
        const float dii = dis[row];
        const float hv  = c[i];
        h_out  [(size_t)row * FOUT + col] = hv;
        agg_out[(size_t)row * FOUT + col] = hv * (dii * dii) + bv;
    }
}

// ---------------------------------------------------------------------------
// agg[dst[e]][:] += h[src[e]][:] * dis[src]*dis[dst]
// One wave32 per edge; one b128 (b64 for F=64) gather per lane, then F/32 f32
// atomics. h rows are L2-resident (25.6 MB << 192 MB) -> L2-atomic bound.
// ---------------------------------------------------------------------------
template <int F>
__global__ void edge_scatter_kernel(const float* __restrict__ h,
                                    const int* __restrict__ src,
                                    const int* __restrict__ dst,
                                    const float* __restrict__ dis,
                                    float* __restrict__ agg, int e) {
    int wave = (blockIdx.x * blockDim.x + threadIdx.x) >> 5;
    int lane = threadIdx.x & 31;
    if (wave >= e) return;
    int s = src[wave];
    int d = dst[wave];
    float norm = dis[s] * dis[d];
    const float* hrow = h + (size_t)s * F;
    float* arow = agg + (size_t)d * F;
    if (F == 128) {
        const float4 v = ((const float4*)hrow)[lane];   // global_load_b128
        float* p = arow + lane * 4;
        atomicAdd(p + 0, v.x * norm);
        atomicAdd(p + 1, v.y * norm);
        atomicAdd(p + 2, v.z * norm);
        atomicAdd(p + 3, v.w * norm);
    } else { // F == 64
        const float2 v = ((const float2*)hrow)[lane];   // global_load_b64
        float* p = arow + lane * 2;
        atomicAdd(p + 0, v.x * norm);
        atomicAdd(p + 1, v.y * norm);
    }
}

// ---------------------------------------------------------------------------
// Host-side launch
// ---------------------------------------------------------------------------
template <int FOUT, bool RELU>
static inline void run_layer(const float* in_feat, const float* W, const float* b,
                             const int* src, const int* dst, const float* dis,
                             float* h_buf, float* agg_out, hipStream_t stream) {
    dim3 gblock(32 * (FOUT / 16));
    dim3 ggrid(GCN_N / 16);            // 3125 exact row tiles
    gemm_wmma_fused_kernel<FOUT, RELU><<<ggrid, gblock, 0, stream>>>(
        in_feat, W, b, dis, h_buf, agg_out);
    edge_scatter_kernel<FOUT><<<(GCN_E + 7) / 8, 256, 0, stream>>>(
        h_buf, src, dst, dis, agg_out, GCN_E);
}

extern "C" void kernel_launch(void* const* d_in, const int* in_sizes, int n_in,
                              void* d_out, int out_size, void* d_ws, size_t ws_size,
                              hipStream_t stream) {
    const float* x   = (const float*)d_in[0];
    const int*   ei  = (const int*)d_in[1];
    const float* W1  = (const float*)d_in[2];
    const float* b1  = (const float*)d_in[3];
    const float* W2  = (const float*)d_in[4];
    const float* b2  = (const float*)d_in[5];
    const float* W3  = (const float*)d_in[6];
    const float* b3  = (const float*)d_in[7];
    const float* W4  = (const float*)d_in[8];
    const float* b4  = (const float*)d_in[9];

    const int* src = ei;           // edge_index[0]
    const int* dst = ei + GCN_E;   // edge_index[1]

    // workspace layout: dis[N] | bufA[N*128] | bufB[N*128]
    float* dis  = (float*)d_ws;
    float* bufA = dis + 65536;                 // 256 KB offset, keeps 16B alignment
    float* bufB = bufA + (size_t)GCN_N * GCN_H;
    float* outf = (float*)d_out;

    // degree normalization dis = rsqrt(1 + indeg)
    deg_init_kernel <<<(GCN_N + 255) / 256, 256, 0, stream>>>(dis, GCN_N);
    deg_count_kernel<<<(GCN_E + 255) / 256, 256, 0, stream>>>(dst, dis, GCN_E);
    deg_rsqrt_kernel<<<(GCN_N + 255) / 256, 256, 0, stream>>>(dis, GCN_N);

    // layer 1: x -> h1=bufA, agg1-init=bufB; scatter -> bufB
    run_layer<GCN_H,  false>(x,    W1, b1, src, dst, dis, bufA, bufB, stream);
    // layer 2: relu(agg1)=bufB -> h2=bufA, agg2-init=bufB (in-place, barrier-safe)
    run_layer<GCN_H,  true >(bufB, W2, b2, src, dst, dis, bufA, bufB, stream);
    // layer 3
    run_layer<GCN_H,  true >(bufB, W3, b3, src, dst, dis, bufA, bufB, stream);
    // layer 4: relu(agg3)=bufB -> h4=bufA (64 wide), agg4-init=d_out; no output relu
    run_layer<GCN_OUT,true >(bufB, W4, b4, src, dst, dis, bufA, outf, stream);
}